// RecurrentParamsComplex_25598005084833
// MI455X (gfx1250) — compile-verified
//
#include <hip/hip_runtime.h>

// CDNA5 / gfx1250, wave32. Batch-in-N WMMA mapping for a 5-wide MLP chain.
// Two 16-row tiles per wave: lanes 0-15 = tile A rows, lanes 16-31 = tile B rows.
typedef float v2f __attribute__((ext_vector_type(2)));
typedef float v8f __attribute__((ext_vector_type(8)));

// ---- raw CDNA5 transcendentals (args are positive, O(1) by construction) ----
#define LN2     0.69314718055994531f   // ln(2)
#define INV2PI  0.15915494309189535f   // 1/(2*pi)  (v_sin/v_cos take turns)
#define LN2_2PI 0.11031860767402886f   // ln(2)/(2*pi)

__device__ __forceinline__ float fast_ln(float v) {           // ln(v)
    return __builtin_amdgcn_logf(v) * LN2;
}
__device__ __forceinline__ float fast_sinln(float v) {        // sin(ln(v))
    return __builtin_amdgcn_sinf(__builtin_amdgcn_logf(v) * LN2_2PI);
}
__device__ __forceinline__ float fast_x_plus_lntan(float v) { // v + ln(tan(v))
    const float t = v * INV2PI;
    const float s = __builtin_amdgcn_sinf(t);
    const float c = __builtin_amdgcn_cosf(t);
    const float d = __builtin_amdgcn_logf(s) - __builtin_amdgcn_logf(c);
    return fmaf(LN2, d, v);               // ln(tan v) = (log2 s - log2 c)*ln2
}

// Exchange wave halves: lane i <-> lane i^16 (ds_swizzle group-of-32, xor=0x10).
__device__ __forceinline__ float swap16(float v) {
    return __int_as_float(
        __builtin_amdgcn_ds_swizzle(__float_as_int(v), 0x401F));
}

// Per-layer constant state: A-matrix (W, M=out-feature in lanes, K padded to 8
// across two 16x16x4 WMMAs) and the bias accumulator C. Rows M=5..15 and the
// whole hi half of C are HARD ZEROS -> accumulator lanes 16-31 are exactly 0,
// which makes half-merges a plain add (no selects, no divergent branches).
struct FcW { v2f a1; v2f a2; v8f c; };

__device__ __forceinline__ FcW make_fcw(const float* __restrict__ W,
                                        const float* __restrict__ b,
                                        int lane) {
    const int  m  = lane & 15;
    const bool hi = lane >= 16;
    const bool mv = (m < 5);
    FcW w;
    // A layout (16x4 f32): V0 = K0(lanes0-15)/K2(lanes16-31), V1 = K1/K3.
    w.a1[0] = mv ? W[m * 5 + (hi ? 2 : 0)] : 0.0f;
    w.a1[1] = mv ? W[m * 5 + (hi ? 3 : 1)] : 0.0f;
    // Second WMMA covers K=4 only; K=5..7 are hard zeros in A.
    w.a2[0] = (mv && !hi) ? W[m * 5 + 4] : 0.0f;
    w.a2[1] = 0.0f;
    v8f c;
#pragma unroll
    for (int v = 0; v < 8; ++v) c[v] = (!hi && v < 5) ? b[v] : 0.0f;
    w.c = c;
    return w;
}

// One affine layer on a 32-row packed wave. g0..g4: feature k of row(lane).
// Produces D accumulators for both tiles (feature v in VGPR v, lanes 0-15;
// lanes 16-31 exactly zero).
__device__ __forceinline__ void fc_tiles(const FcW& w, v8f cin,
                                         float g0, float g1, float g2, float g3,
                                         float g4, bool hi, v8f& dA, v8f& dB) {
    const float s0 = swap16(g0), s1 = swap16(g1), s2 = swap16(g2),
                s3 = swap16(g3), s4 = swap16(g4);
    // B layout (4x16 f32): V0 = K0(lanes0-15)/K2(lanes16-31), V1 = K1/K3.
    v2f bA1, bA2, bB1, bB2;
    bA1[0] = hi ? s2 : g0;  bA1[1] = hi ? s3 : g1;
    bB1[0] = hi ? g2 : s0;  bB1[1] = hi ? g3 : s1;
    bA2[0] = g4;  bA2[1] = 0.0f;   // hi-lane K=6 slot hits zero A column
    bB2[0] = s4;  bB2[1] = 0.0f;
    dA = __builtin_amdgcn_wmma_f32_16x16x4_f32(false, w.a1, false, bA1,
                                               (short)0, cin, false, false);
    dB = __builtin_amdgcn_wmma_f32_16x16x4_f32(false, w.a1, false, bB1,
                                               (short)0, cin, false, false);
    dA = __builtin_amdgcn_wmma_f32_16x16x4_f32(false, w.a2, false, bA2,
                                               (short)0, dA, false, false);
    dB = __builtin_amdgcn_wmma_f32_16x16x4_f32(false, w.a2, false, bB2,
                                               (short)0, dB, false, false);
}

// Merge tiles back to packed per-lane form: dA is zero in lanes 16-31, and
// swap16(dB) is zero in lanes 0-15, so the merge is a single unconditional add.
#define FC(W, CIN) do { v8f dA, dB;                                         \
        fc_tiles((W), (CIN), f0, f1, f2, f3, f4, hi, dA, dB);               \
        f0 = dA[0] + swap16(dB[0]);  f1 = dA[1] + swap16(dB[1]);            \
        f2 = dA[2] + swap16(dB[2]);  f3 = dA[3] + swap16(dB[3]);            \
        f4 = dA[4] + swap16(dB[4]); } while (0)

#define MAP5(OP)   do { f0 = OP(f0); f1 = OP(f1); f2 = OP(f2); \
                        f3 = OP(f3); f4 = OP(f4); } while (0)
#define AFF5(a, c) do { f0 = fmaf((a), f0, (c)); f1 = fmaf((a), f1, (c)); \
                        f2 = fmaf((a), f2, (c)); f3 = fmaf((a), f3, (c)); \
                        f4 = fmaf((a), f4, (c)); } while (0)

__global__ __launch_bounds__(256) void mlp_chain_wmma(
    const float* __restrict__ x,
    const float* __restrict__ W1, const float* __restrict__ b1,
    const float* __restrict__ W2, const float* __restrict__ b2,
    float* __restrict__ out, long long nrows) {
    const int  lane = (int)(threadIdx.x & 31u);
    const bool hi   = lane >= 16;
    const long long wave = (long long)blockIdx.x * (blockDim.x >> 5)
                         + (threadIdx.x >> 5);
    const long long row   = wave * 32 + lane;                 // 32 rows / wave
    const long long ldrow = (row < nrows) ? row : (nrows - 1);  // keep EXEC full

    const FcW w1 = make_fcw(W1, b1, lane);
    const FcW w2 = make_fcw(W2, b2, lane);
    const v8f c2x2 = w2.c + w2.c;  // bias for fused fc2(x)+fc2(z)

    // Each lane owns one row: 640B contiguous per wave, no duplicate addresses.
    const float* xr = x + ldrow * 5;
    float f0 = xr[0] + 1.0f, f1 = xr[1] + 1.0f, f2 = xr[2] + 1.0f,
          f3 = xr[3] + 1.0f, f4 = xr[4] + 1.0f;           // x = x + 1

    FC(w1, w1.c);  AFF5(2.0f, 0.0f);                      // fc1; x *= 2
    FC(w2, w2.c);  MAP5(fast_sinln);                      // fc2; sin(log(x))
    FC(w2, w2.c);  MAP5(fast_ln);                         // fc2; log
    FC(w1, w1.c);  AFF5(4.0f, 6.0f);                      // fc1; ((x+1)*2+1)*2
    FC(w2, w2.c);  MAP5(fast_ln);                         // fc2; log
    FC(w1, w1.c);  AFF5(1.0f, 2.0f);                      // fc1; x + 2
    FC(w2, w2.c);  MAP5(fast_ln);                         // fc2; log
    FC(w2, w2.c);  MAP5(fast_ln);                         // fc2; log
    FC(w1, w1.c);  AFF5(2.0f, 2.0f);                      // fc1; (x+1)*2
    FC(w2, w2.c);                                         // fc2 -> final X in f*

    // y = fc1(sin(log(X))) + 3
    const float y0 = fast_sinln(f0), y1 = fast_sinln(f1), y2 = fast_sinln(f2),
                y3 = fast_sinln(f3), y4 = fast_sinln(f4);
    v8f dYA, dYB;
    fc_tiles(w1, w1.c, y0, y1, y2, y3, y4, hi, dYA, dYB);

    // z = fc2(X) + fc2(log(tan(X))) = fc2(X + log(tan(X))) with bias 2*b2
    const float z0 = fast_x_plus_lntan(f0);
    const float z1 = fast_x_plus_lntan(f1);
    const float z2 = fast_x_plus_lntan(f2);
    const float z3 = fast_x_plus_lntan(f3);
    const float z4 = fast_x_plus_lntan(f4);
    v8f dZA, dZB;
    fc_tiles(w2, c2x2, z0, z1, z2, z3, z4, hi, dZA, dZB);

    // a = (y + 3) * z per tile: hi halves are (0+3)*0 == 0, so the merge is
    // again a plain add of the swapped tile-B product. Lane k stores row base+k.
    const float a0 = (dYA[0] + 3.0f) * dZA[0] + swap16((dYB[0] + 3.0f) * dZB[0]);
    const float a1 = (dYA[1] + 3.0f) * dZA[1] + swap16((dYB[1] + 3.0f) * dZB[1]);
    const float a2 = (dYA[2] + 3.0f) * dZA[2] + swap16((dYB[2] + 3.0f) * dZB[2]);
    const float a3 = (dYA[3] + 3.0f) * dZA[3] + swap16((dYB[3] + 3.0f) * dZB[3]);
    const float a4 = (dYA[4] + 3.0f) * dZA[4] + swap16((dYB[4] + 3.0f) * dZB[4]);

    if (row < nrows) {
        float* o = out + row * 5;
        o[0] = a0; o[1] = a1; o[2] = a2; o[3] = a3; o[4] = a4;
    }
}

extern "C" void kernel_launch(void* const* d_in, const int* in_sizes, int n_in,
                              void* d_out, int out_size, void* d_ws, size_t ws_size,
                              hipStream_t stream) {
    const float* x  = (const float*)d_in[0];
    const float* W1 = (const float*)d_in[1];
    const float* b1 = (const float*)d_in[2];
    const float* W2 = (const float*)d_in[3];
    const float* b2 = (const float*)d_in[4];
    float* out = (float*)d_out;

    const long long nrows = (long long)in_sizes[0] / 5;   // 4,194,304
    const int block = 256;                                // 8 waves, 256 rows/block
    const long long rows_per_block = (block / 32) * 32;
    const int grid = (int)((nrows + rows_per_block - 1) / rows_per_block);

    mlp_chain_wmma<<<grid, block, 0, stream>>>(x, W1, b1, W2, b2, out, nrows);
}